// BahdanauAttentionVec_29085518529187
// MI455X (gfx1250) — compile-verified
//
#include <hip/hip_runtime.h>

typedef __attribute__((ext_vector_type(2))) float v2f;
typedef __attribute__((ext_vector_type(4))) float v4f;
typedef __attribute__((ext_vector_type(8))) float v8f;
typedef __attribute__((ext_vector_type(4))) unsigned int u32x4;
typedef __attribute__((ext_vector_type(8))) int i32x8;
typedef __attribute__((ext_vector_type(4))) int i32x4;

#define BATCH 8
#define T_ENC 256
#define T_DEC 256
#define ENC_DIM 512
#define UNITS 256

// 2*log2(e): projections are stored pre-scaled so the score loop can use
// exp2 directly:  tanh(x) = 1 - 2/(exp2(C*x)+1)  with C = 2*log2(e).
#define TANH_SCALE 2.8853900817779268f

// ---------------------------------------------------------------------------
// Kernel 1: P = (X (R x K) * W (K x 256) + bias) * scale, R = 2048 (= B*T).
// One wave per 16x16 tile, V_WMMA_F32_16X16X4_F32, K loop unrolled by 8.
// TRANSPOSE=1 stores out[(b*256+n)*256 + t]  (peT layout (B, U, T_enc)),
// TRANSPOSE=0 stores out[r*256 + n]          (pd  layout (B, T_dec, U)).
// ---------------------------------------------------------------------------
template <int K, int TRANSPOSE>
__global__ __launch_bounds__(256) void gemm_wmma(const float* __restrict__ X,
                                                 const float* __restrict__ W,
                                                 const float* __restrict__ bias,
                                                 float* __restrict__ out,
                                                 float scale) {
  const int tid  = threadIdx.x;
  const int wave = tid >> 5;
  const int lane = tid & 31;
  const int half = lane >> 4;   // K-half selector per WMMA f32 A/B layout
  const int l15  = lane & 15;

  const int tile  = blockIdx.x * 8 + wave;  // 0..2047 (128 M-tiles x 16 N-tiles)
  const int tileM = tile >> 4;
  const int tileN = tile & 15;

  const int rowA = tileM * 16 + l15;        // both lane halves cover M=0..15
  const int colB = tileN * 16 + l15;
  const float* ap = X + (size_t)rowA * K + half * 2;     // A: K = k + half*2 + {0,1}
  const float* bp = W + (size_t)(half * 2) * 256 + colB; // B: rows k+half*2+{0,1}

  v8f c = {};
#pragma unroll 8
  for (int k = 0; k < K; k += 4) {
    v2f a, bv;
    a.x  = ap[k];
    a.y  = ap[k + 1];
    bv.x = bp[(size_t)k * 256];
    bv.y = bp[(size_t)k * 256 + 256];
    c = __builtin_amdgcn_wmma_f32_16x16x4_f32(false, a, false, bv,
                                              (short)0, c, false, false);
  }

  // D layout: VGPR r -> M = r + 8*half, N = l15
#pragma unroll
  for (int r = 0; r < 8; ++r) {
    const int m    = r + half * 8;
    const int gRow = tileM * 16 + m;       // = b*256 + t
    const int n    = tileN * 16 + l15;     // u index
    const float val = (c[r] + bias[n]) * scale;
    if (TRANSPOSE) {
      const int bb = gRow >> 8;
      const int t  = gRow & 255;
      out[((size_t)bb * 256 + n) * 256 + t] = val;
    } else {
      out[(size_t)gRow * 256 + n] = val;
    }
  }
}

// ---------------------------------------------------------------------------
// Kernel 2: fused score + softmax + context.
// Block = (b, s-group of 16). peT[b] tile (256 KB) is DMA'd into LDS by the
// Tensor Data Mover and reused for 16 decoder rows (CDNA5 320 KB WGP LDS).
// ---------------------------------------------------------------------------
__global__ __launch_bounds__(256) void attn_kernel(const float* __restrict__ enc,
                                                   const float* __restrict__ peT,
                                                   const float* __restrict__ pd,
                                                   const float* __restrict__ Vw,
                                                   const float* __restrict__ Vb,
                                                   float* __restrict__ ctx_out,
                                                   float* __restrict__ attn_out) {
  extern __shared__ float smem[];
  float* pe_s     = smem;                  // 65536 floats (256 KB): peT[b][u][t]
  float* attn_all = pe_s + 65536;          // 16*256
  float* pd_s     = attn_all + 4096;       // 256
  float* vw_s     = pd_s + 256;            // 256
  float* red_s    = vw_s + 256;            // 16

  const int j    = threadIdx.x;            // 0..255
  const int lane = j & 31;
  const int wv   = j >> 5;
  const int b    = blockIdx.y;             // 0..7
  const int sg   = blockIdx.x;             // 0..15

  const float* peb = peT + (size_t)b * UNITS * T_ENC;

  // --- TDM: async-copy the 256 KB pe tile into LDS (wave 0 issues) ---------
  if (wv == 0) {
    const unsigned long long ga = (unsigned long long)(size_t)peb;
    const unsigned int ldsOff   = (unsigned int)(size_t)pe_s;
    // D# group 0: count=1 | lds_addr | global_addr[31:0] | ga[56:32], type=2
    u32x4 g0 = {1u, ldsOff, (unsigned int)ga,
                ((unsigned int)(ga >> 32) & 0x01FFFFFFu) | (2u << 30)};
    // D# group 1: 1-D copy of 32768 x 8-byte elements (data_size=3)
    i32x8 g1 = {(int)(3u << 16),        // data_size = 8B
                (int)(32768u << 16),    // tensor_dim0 = 32768 (bits 63:48)
                (int)(1u << 16),        // tensor_dim1 = 1     (bits 95:80)
                (int)(32768u << 16),    // tile_dim0  = 32768  (bits 127:112)
                1,                      // tile_dim1  = 1      (bits 143:128)
                32768,                  // tensor_dim0_stride  (bits 191:160)
                0, 0};
    i32x4 gz4 = {0, 0, 0, 0};
    i32x8 gz8 = {0, 0, 0, 0, 0, 0, 0, 0};
    __builtin_amdgcn_tensor_load_to_lds(g0, g1, gz4, gz4, gz8, 0);
    __builtin_amdgcn_s_wait_tensorcnt(0);
  }
  vw_s[j] = Vw[j];
  const float vb = Vb[0];
  __syncthreads();

  // sum of Vw (for score = vwsum - 2*acc + vb), computed once per block
  float vwsum = 0.0f;
#pragma unroll 4
  for (int u = 0; u < 256; u += 4) {
    const v4f w4 = *(const v4f*)&vw_s[u];
    vwsum += (w4.x + w4.y) + (w4.z + w4.w);
  }
  const float base = vwsum + vb;

  for (int sl = 0; sl < 16; ++sl) {
    const int s = sg * 16 + sl;
    pd_s[j] = pd[((size_t)b * 256 + s) * 256 + j];
    __syncthreads();

    // score for t = j.  pe/pd are pre-scaled by 2*log2(e):
    //   acc = sum_u vw[u] / (exp2(pd'+pe') + 1);  score = base - 2*acc
    float acc = 0.0f;
    for (int u = 0; u < 256; u += 4) {
      const v4f pd4 = *(const v4f*)&pd_s[u];      // ds_load_b128 broadcast
      const v4f vw4 = *(const v4f*)&vw_s[u];
#pragma unroll
      for (int q = 0; q < 4; ++q) {
        const float e = __builtin_amdgcn_exp2f(pd4[q] + pe_s[(u + q) * 256 + j]);
        acc = fmaf(vw4[q], __builtin_amdgcn_rcpf(e + 1.0f), acc);
      }
    }
    const float score = fmaf(-2.0f, acc, base);

    // softmax over t: wave shuffle reduction + 8-entry cross-wave pass
    float mx = score;
#pragma unroll
    for (int m = 16; m > 0; m >>= 1) mx = fmaxf(mx, __shfl_xor(mx, m, 32));
    if (lane == 0) red_s[wv] = mx;
    __syncthreads();
    mx = red_s[0];
#pragma unroll
    for (int q = 1; q < 8; ++q) mx = fmaxf(mx, red_s[q]);

    const float e = __builtin_amdgcn_exp2f((score - mx) * 1.4426950408889634f);
    float sum = e;
#pragma unroll
    for (int m = 16; m > 0; m >>= 1) sum += __shfl_xor(sum, m, 32);
    if (lane == 0) red_s[8 + wv] = sum;
    __syncthreads();
    sum = red_s[8];
#pragma unroll
    for (int q = 1; q < 8; ++q) sum += red_s[8 + q];

    const float w = e * __builtin_amdgcn_rcpf(sum);
    attn_all[sl * 256 + j] = w;
    attn_out[((size_t)b * 256 + s) * 256 + j] = w;
    __syncthreads();   // protects pd_s / red_s for the next decoder row
  }

  // context: thread j owns e = j and e = j+256; 16 s-rows in registers so
  // enc[b] (512 KB) streams from L2 exactly once per block.
  float accA[16], accB[16];
#pragma unroll
  for (int sl = 0; sl < 16; ++sl) { accA[sl] = 0.0f; accB[sl] = 0.0f; }

  const float* encb = enc + (size_t)b * T_ENC * ENC_DIM;
  for (int t = 0; t < 256; t += 4) {
    __builtin_prefetch(&encb[(size_t)((t + 16) & 255) * 512 + j], 0, 1);
    float e0[4], e1[4];
#pragma unroll
    for (int q = 0; q < 4; ++q) {
      e0[q] = encb[(size_t)(t + q) * 512 + j];
      e1[q] = encb[(size_t)(t + q) * 512 + j + 256];
    }
#pragma unroll
    for (int sl = 0; sl < 16; ++sl) {
      const v4f w4 = *(const v4f*)&attn_all[sl * 256 + t]; // b128 broadcast
      accA[sl] = fmaf(w4.x, e0[0], accA[sl]);
      accA[sl] = fmaf(w4.y, e0[1], accA[sl]);
      accA[sl] = fmaf(w4.z, e0[2], accA[sl]);
      accA[sl] = fmaf(w4.w, e0[3], accA[sl]);
      accB[sl] = fmaf(w4.x, e1[0], accB[sl]);
      accB[sl] = fmaf(w4.y, e1[1], accB[sl]);
      accB[sl] = fmaf(w4.z, e1[2], accB[sl]);
      accB[sl] = fmaf(w4.w, e1[3], accB[sl]);
    }
  }

#pragma unroll
  for (int sl = 0; sl < 16; ++sl) {
    const int s = sg * 16 + sl;
    ctx_out[((size_t)b * 256 + s) * 512 + j]       = accA[sl];
    ctx_out[((size_t)b * 256 + s) * 512 + j + 256] = accB[sl];
  }
}

extern "C" void kernel_launch(void* const* d_in, const int* in_sizes, int n_in,
                              void* d_out, int out_size, void* d_ws, size_t ws_size,
                              hipStream_t stream) {
  (void)in_sizes; (void)n_in; (void)out_size; (void)ws_size;
  const float* enc = (const float*)d_in[0];
  const float* dec = (const float*)d_in[1];
  const float* W1  = (const float*)d_in[2];
  const float* b1  = (const float*)d_in[3];
  const float* W2  = (const float*)d_in[4];
  const float* b2  = (const float*)d_in[5];
  const float* Vw  = (const float*)d_in[6];
  const float* Vb  = (const float*)d_in[7];

  float* peT = (float*)d_ws;                              // (B,U,T_enc) 2 MB
  float* pd  = peT + (size_t)BATCH * UNITS * T_ENC;       // (B,T_dec,U) 2 MB

  float* ctx  = (float*)d_out;                            // (B,T_dec,512)
  float* attn = ctx + (size_t)BATCH * T_DEC * ENC_DIM;    // (B,T_dec,256)

  // Projections via FP32 WMMA, pre-scaled by 2*log2(e) for the tanh rewrite.
  gemm_wmma<ENC_DIM, 1><<<256, 256, 0, stream>>>(enc, W1, b1, peT, TANH_SCALE);
  gemm_wmma<UNITS,   0><<<256, 256, 0, stream>>>(dec, W2, b2, pd, TANH_SCALE);

  const size_t shbytes = (size_t)(65536 + 4096 + 256 + 256 + 16) * sizeof(float);
  (void)hipFuncSetAttribute((const void*)attn_kernel,
                            hipFuncAttributeMaxDynamicSharedMemorySize,
                            (int)shbytes);
  attn_kernel<<<dim3(16, 8), 256, shbytes, stream>>>(enc, peT, pd, Vw, Vb, ctx, attn);
}